// Qwen25_SigLIPAttention_73203422593365
// MI455X (gfx1250) — compile-verified
//
#include <hip/hip_runtime.h>
#include <stdint.h>

// Problem constants (match reference)
#define H_    16
#define KVH_  4
#define S_    2048
#define D_    128
#define BM    128            // query rows per workgroup (8 waves x 16 rows)
#define WAVES 8
#define BN    64             // key block (columns) per iteration
#define NBLK  (S_ / BN)      // 32
#define SCALE_QK 0.08838834764831845f   // 1/sqrt(128)
#define MIN_VAL  (-1.0e7f)

typedef __attribute__((ext_vector_type(2))) float        v2f;
typedef __attribute__((ext_vector_type(8))) float        v8f;
typedef __attribute__((ext_vector_type(4))) unsigned int u32x4;
typedef __attribute__((ext_vector_type(8))) int          i32x8;
typedef __attribute__((ext_vector_type(4))) int          i32x4;

// LDS layout (bytes):
//   [0      ..  32768) K buffer 0   (64x128 f32)
//   [32768  ..  65536) K buffer 1
//   [65536  ..  98304) V buffer 0
//   [98304  .. 131072) V buffer 1
//   [131072 .. 163840) P staging (8 waves x 16x64 f32)
#define K_OFF(b)  ((unsigned)(b) * 32768u)
#define V_OFF(b)  (65536u + (unsigned)(b) * 32768u)
#define P_OFF     131072u
#define LDS_BYTES 163840u

// ---- Tensor Data Mover: DMA a 64x128 fp32 tile (contiguous rows, row stride
// 128 elements) from global memory into LDS.  Descriptor per CDNA5 ISA §8.
__device__ __forceinline__ void tdm_load_64x128(uint32_t lds_byte_off,
                                                const float* gsrc) {
  uint64_t ga = (uint64_t)(uintptr_t)gsrc;
  u32x4 g0;
  g0[0] = 1u;                                   // count=1 (valid), no gather
  g0[1] = lds_byte_off;                         // lds_addr
  g0[2] = (uint32_t)ga;                         // global_addr[31:0]
  g0[3] = ((uint32_t)(ga >> 32) & 0x01FFFFFFu)  // global_addr[56:32]
          | (2u << 30);                         // type = 2 ("image")
  i32x8 g1;
  g1[0] = (int)(2u << 16);                      // data_size = 4 bytes
  g1[1] = (int)((uint32_t)D_ << 16);            // tensor_dim0 = 128 (bits 63:48)
  g1[2] = (int)((uint32_t)S_ << 16);            // tensor_dim1 lo16 (bits 95:80)
  g1[3] = (int)((uint32_t)D_ << 16);            // tensor_dim1 hi16=0, tile_dim0 = 128
  g1[4] = (int)(uint32_t)BN;                    // tile_dim1 = 64, tile_dim2 = 0
  g1[5] = (int)(uint32_t)D_;                    // tensor_dim0_stride = 128 elements
  g1[6] = 0;                                    // stride hi, dim1_stride lo
  g1[7] = 0;
  i32x4 z4 = {0, 0, 0, 0};                      // groups 2/3 unused (2-D tensor)
  i32x8 z8 = {0, 0, 0, 0, 0, 0, 0, 0};          // 6-arg form: extra group, zero
  __builtin_amdgcn_tensor_load_to_lds(g0, g1, z4, z4, z8, 0);
}

__global__ __launch_bounds__(256, 1)
void dual_path_flash_attn(const float* __restrict__ Qs,
                          const float* __restrict__ Qd,
                          const float* __restrict__ K,
                          const float* __restrict__ V,
                          const unsigned char* __restrict__ inter_mask,
                          const unsigned char* __restrict__ intra_mask,
                          float* __restrict__ Out) {
  extern __shared__ char smem_raw[];
  float* smem = (float*)smem_raw;
  const uint32_t lds_base = (uint32_t)(uintptr_t)smem_raw;  // LDS byte offset

  const int tid  = threadIdx.x;
  const int lane = tid & 31;
  const int wid  = tid >> 5;
  const int lg   = lane >> 4;      // lane group (0: lanes 0-15, 1: lanes 16-31)
  const int ln   = lane & 15;

  const int h     = blockIdx.x / (S_ / BM);
  const int qtile = blockIdx.x % (S_ / BM);
  const int kvh   = h / (H_ / KVH_);
  const int qbase = qtile * BM + wid * 16;     // first query row of this wave

  const float* Kh  = K  + (size_t)kvh * S_ * D_;
  const float* Vh  = V  + (size_t)kvh * S_ * D_;
  const float* Qsh = Qs + (size_t)h * S_ * D_;
  const float* Qdh = Qd + (size_t)h * S_ * D_;

  // ---- preload Q fragments (A-operand layout of V_WMMA_F32_16X16X4_F32):
  // lane L: row m = L%16, VGPR pair = elements [4*kk + 2*(L/16), +1]
  v2f qs[32], qd[32];
  {
    const size_t qrow = (size_t)(qbase + ln) * D_;
    #pragma unroll
    for (int kk = 0; kk < 32; ++kk) {
      const int dcol = 4 * kk + 2 * lg;
      qs[kk] = *(const v2f*)(Qsh + qrow + dcol);
      qd[kk] = *(const v2f*)(Qdh + qrow + dcol);
    }
  }

  const v8f zero8 = {0.f, 0.f, 0.f, 0.f, 0.f, 0.f, 0.f, 0.f};
  v8f acc[8];                       // 16x128 output accumulator (8 C tiles)
  #pragma unroll
  for (int dt = 0; dt < 8; ++dt) acc[dt] = zero8;

  float m_run[8], l_run[8];         // row stats; row = r + 8*lg (C layout)
  #pragma unroll
  for (int r = 0; r < 8; ++r) { m_run[r] = -3.0e38f; l_run[r] = 0.0f; }

  float* Plds = smem + (P_OFF / 4) + wid * (16 * BN);   // wave-private P tile

  // ---- prefetch block 0 (K and V) via TDM, wave 0 only ----
  if (wid == 0) {
    tdm_load_64x128(lds_base + K_OFF(0), Kh);
    tdm_load_64x128(lds_base + V_OFF(0), Vh);
  }

  for (int j = 0; j < NBLK; ++j) {
    const int buf = j & 1;
    if (wid == 0) {
      if (j + 1 < NBLK) {     // prefetch next block into the other buffer
        tdm_load_64x128(lds_base + K_OFF(1 - buf), Kh + (size_t)(j + 1) * BN * D_);
        tdm_load_64x128(lds_base + V_OFF(1 - buf), Vh + (size_t)(j + 1) * BN * D_);
        __builtin_amdgcn_s_wait_tensorcnt(2);   // block j's pair has landed
      } else {
        __builtin_amdgcn_s_wait_tensorcnt(0);
      }
    }
    __syncthreads();

    const float* Klds = smem + K_OFF(buf) / 4;
    const float* Vlds = smem + V_OFF(buf) / 4;

    // ---- dual-path scores: 4 tiles of 16x16, K-dim = 128 in steps of 4 ----
    v8f ss[4], sd[4];
    #pragma unroll
    for (int t = 0; t < 4; ++t) { ss[t] = zero8; sd[t] = zero8; }
    #pragma unroll
    for (int t = 0; t < 4; ++t) {
      #pragma unroll
      for (int kk = 0; kk < 32; ++kk) {
        // B-operand: B[k][n] = K[n][d]; lane L: n = L%16, k = 2*(L/16)+{0,1}
        v2f b = *(const v2f*)(Klds + (t * 16 + ln) * D_ + 4 * kk + 2 * lg);
        ss[t] = __builtin_amdgcn_wmma_f32_16x16x4_f32(
            false, qs[kk], false, b, (short)0, ss[t], false, false);
        sd[t] = __builtin_amdgcn_wmma_f32_16x16x4_f32(
            false, qd[kk], false, b, (short)0, sd[t], false, false);
      }
    }

    // ---- mask select (intra ? dynamic : static), scale, keep-fill ----
    const int kbase = j * BN;
    #pragma unroll
    for (int t = 0; t < 4; ++t) {
      const int kcol = kbase + t * 16 + ln;
      #pragma unroll
      for (int r = 0; r < 8; ++r) {
        const int q = qbase + r + 8 * lg;
        const size_t mi = (size_t)q * S_ + kcol;
        const unsigned char mA = intra_mask[mi];
        const unsigned char mE = inter_mask[mi];
        float sv = (mA ? sd[t][r] : ss[t][r]) * SCALE_QK;
        ss[t][r] = (mA | mE) ? sv : MIN_VAL;     // reuse ss as selected score
      }
    }

    // ---- online softmax: per-row max over 64 columns ----
    float csc[8], rsum[8];
    #pragma unroll
    for (int r = 0; r < 8; ++r) {
      float v = fmaxf(fmaxf(ss[0][r], ss[1][r]), fmaxf(ss[2][r], ss[3][r]));
      #pragma unroll
      for (int off = 1; off < 16; off <<= 1)   // stays within 16-lane group
        v = fmaxf(v, __shfl_xor(v, off, 32));
      const float mn = fmaxf(m_run[r], v);
      csc[r]  = __expf(m_run[r] - mn);
      m_run[r] = mn;
      rsum[r] = 0.f;
    }

    // ---- exponentiate, row-sum, stage P into LDS (C-layout -> dense) ----
    #pragma unroll
    for (int t = 0; t < 4; ++t) {
      #pragma unroll
      for (int r = 0; r < 8; ++r) {
        const float p = __expf(ss[t][r] - m_run[r]);
        rsum[r] += p;
        Plds[(r + 8 * lg) * BN + t * 16 + ln] = p;
      }
    }
    #pragma unroll
    for (int r = 0; r < 8; ++r) {
      float v = rsum[r];
      #pragma unroll
      for (int off = 1; off < 16; off <<= 1)
        v += __shfl_xor(v, off, 32);
      l_run[r] = l_run[r] * csc[r] + v;
    }
    #pragma unroll
    for (int dt = 0; dt < 8; ++dt)
      #pragma unroll
      for (int r = 0; r < 8; ++r)
        acc[dt][r] *= csc[r];

    // ---- P @ V : K-dim = 64 in steps of 4, 8 output d-tiles ----
    #pragma unroll
    for (int kk = 0; kk < 16; ++kk) {
      v2f a = *(const v2f*)(Plds + ln * BN + 4 * kk + 2 * lg);
      #pragma unroll
      for (int dt = 0; dt < 8; ++dt) {
        v2f b;
        b.x = Vlds[(4 * kk + 2 * lg) * D_ + dt * 16 + ln];
        b.y = Vlds[(4 * kk + 2 * lg + 1) * D_ + dt * 16 + ln];
        acc[dt] = __builtin_amdgcn_wmma_f32_16x16x4_f32(
            false, a, false, b, (short)0, acc[dt], false, false);
      }
    }
    __syncthreads();   // buffer buf may be overwritten next iteration
  }

  // ---- epilogue: normalize and store (B, S, H, D) ----
  #pragma unroll
  for (int r = 0; r < 8; ++r) {
    const float inv = 1.0f / l_run[r];
    const size_t q = (size_t)(qbase + r + 8 * lg);
    #pragma unroll
    for (int dt = 0; dt < 8; ++dt)
      Out[(q * H_ + h) * D_ + dt * 16 + ln] = acc[dt][r] * inv;
  }
}

extern "C" void kernel_launch(void* const* d_in, const int* in_sizes, int n_in,
                              void* d_out, int out_size, void* d_ws, size_t ws_size,
                              hipStream_t stream) {
  (void)in_sizes; (void)n_in; (void)out_size; (void)d_ws; (void)ws_size;
  const float* Qs = (const float*)d_in[0];
  const float* Qd = (const float*)d_in[1];
  const float* K  = (const float*)d_in[2];
  const float* V  = (const float*)d_in[3];
  const unsigned char* inter = (const unsigned char*)d_in[4];
  const unsigned char* intra = (const unsigned char*)d_in[5];
  float* out = (float*)d_out;

  static int attr_done = 0;  // idempotent host-side attribute (not a stream op)
  if (!attr_done) {
    (void)hipFuncSetAttribute((const void*)dual_path_flash_attn,
                              hipFuncAttributeMaxDynamicSharedMemorySize,
                              (int)LDS_BYTES);
    attr_done = 1;
  }

  dim3 grid(H_ * (S_ / BM));   // 256 workgroups
  dim3 block(WAVES * 32);      // 8 wave32 waves
  dual_path_flash_attn<<<grid, block, LDS_BYTES, stream>>>(
      Qs, Qd, K, V, inter, intra, out);
}